// GraphConvolutionalLayer_12927851561632
// MI455X (gfx1250) — compile-verified
//
#include <hip/hip_runtime.h>
#include <hip/hip_bf16.h>

typedef __attribute__((ext_vector_type(2))) float v2f;
typedef __attribute__((ext_vector_type(8))) float v8f;

#define N_NODES 100000
#define N_EDGES 625000
#define D_IN    128
#define D_OUT   128
#define N_ETYPES 64

// ---------------------------------------------------------------------------
// Kernel 1: h = node_features @ W + b   (f32 WMMA 16x16x4)
// grid = N_NODES/16 blocks of 256 threads (8 waves). Wave w computes the
// 16x16 tile at rows [blockIdx*16, +16), cols [w*16, +16).
// Writes h into hbuf (scratch, read by edge kernel) AND into out (seeds the
// self-contribution so the scatter kernel can accumulate directly into out).
// ---------------------------------------------------------------------------
__global__ void __launch_bounds__(256)
gemm_wmma_f32(const float* __restrict__ X, const float* __restrict__ W,
              const float* __restrict__ bias,
              float* __restrict__ hbuf, float* __restrict__ out)
{
    const int lane  = threadIdx.x & 31;
    const int wave  = threadIdx.x >> 5;          // 0..7  -> column tile
    const int half  = lane >> 4;                 // 0 or 1
    const int l15   = lane & 15;

    const int row16 = blockIdx.x * 16;           // M tile base
    const int col16 = wave * 16;                 // N tile base

    // A layout (16x4 f32): lanes 0-15 -> K={0,1}, lanes 16-31 -> K={2,3}
    const int arow = row16 + l15;
    const int akb  = half * 2;
    // B layout (4x16 f32): lanes 0-15 -> rows K={0,1}, lanes 16-31 -> K={2,3}
    const int bcol = col16 + l15;

    v8f acc = {};

    const float* __restrict__ aptr = X + (size_t)arow * D_IN + akb;

#pragma unroll 4
    for (int k0 = 0; k0 < D_IN; k0 += 4) {
        v2f a = *(const v2f*)(aptr + k0);                       // 8B load
        v2f bv;
        bv.x = W[(size_t)(k0 + akb + 0) * D_OUT + bcol];
        bv.y = W[(size_t)(k0 + akb + 1) * D_OUT + bcol];
        acc = __builtin_amdgcn_wmma_f32_16x16x4_f32(
                  /*neg_a=*/false, a, /*neg_b=*/false, bv,
                  /*c_mod=*/(short)0, acc,
                  /*reuse_a=*/false, /*reuse_b=*/false);
    }

    // C/D layout: n = lane&15, VGPR r holds m = (lane>>4)*8 + r
    const int   n   = col16 + l15;
    const int   mb  = row16 + half * 8;
    const float bv  = bias[n];

#pragma unroll
    for (int r = 0; r < 8; ++r) {
        const size_t idx = (size_t)(mb + r) * D_OUT + n;
        const float v = acc[r] + bv;
        hbuf[idx] = v;
        out[idx]  = v;   // self contribution; edges accumulate on top
    }
}

// ---------------------------------------------------------------------------
// Kernel 2: per-edge message + scatter-add.
// One wave per edge; each lane owns a float4 chunk of the 128-wide feature.
// Hardware f32 atomics via inline asm -> global_atomic_add_f32.
// ---------------------------------------------------------------------------
__device__ __forceinline__ void gatomic_fadd(float* p, float v) {
    asm volatile("global_atomic_add_f32 %0, %1, off"
                 :: "v"(p), "v"(v) : "memory");
}

__global__ void __launch_bounds__(256)
edge_scatter(const float* __restrict__ h,
             const int*  __restrict__ eidx,   // (2, E) flattened
             const int*  __restrict__ etype,  // (E,)
             const float* __restrict__ emb,   // (N_ETYPES, 128)
             float* __restrict__ out)
{
    const int e    = blockIdx.x * 8 + (threadIdx.x >> 5);   // edge id
    const int lane = threadIdx.x & 31;

    const int src = eidx[e];
    const int tgt = eidx[N_EDGES + e];
    const int et  = etype[e];

    const int c = lane * 4;
    const float4 hv = *(const float4*)(h   + (size_t)src * D_OUT + c);
    const float4 ev = *(const float4*)(emb + (size_t)et  * D_OUT + c);

    float4 m;
    m.x = hv.x + ev.x;  m.y = hv.y + ev.y;
    m.z = hv.z + ev.z;  m.w = hv.w + ev.w;

    float* o = out + (size_t)tgt * D_OUT + c;
    gatomic_fadd(o + 0, m.x);
    gatomic_fadd(o + 1, m.y);
    gatomic_fadd(o + 2, m.z);
    gatomic_fadd(o + 3, m.w);
}

// ---------------------------------------------------------------------------
// Kernel 3: in-place ReLU epilogue.
// ---------------------------------------------------------------------------
__global__ void __launch_bounds__(256)
relu_inplace(float* __restrict__ out)
{
    const size_t i = ((size_t)blockIdx.x * 256 + threadIdx.x) * 4;
    float4 v = *(float4*)(out + i);
    v.x = fmaxf(v.x, 0.f);  v.y = fmaxf(v.y, 0.f);
    v.z = fmaxf(v.z, 0.f);  v.w = fmaxf(v.w, 0.f);
    *(float4*)(out + i) = v;
}

extern "C" void kernel_launch(void* const* d_in, const int* in_sizes, int n_in,
                              void* d_out, int out_size, void* d_ws, size_t ws_size,
                              hipStream_t stream) {
    const float* node_features = (const float*)d_in[0];
    const int*   edge_index    = (const int*)  d_in[1];
    const int*   edge_type     = (const int*)  d_in[2];
    const float* edge_emb      = (const float*)d_in[3];
    const float* W             = (const float*)d_in[4];
    const float* b             = (const float*)d_in[5];

    float* out  = (float*)d_out;
    float* hbuf = (float*)d_ws;                 // N_NODES * D_OUT floats (51.2 MB)

    // Phase 1: h = X@W + b  (also seeds out with h)
    gemm_wmma_f32<<<N_NODES / 16, 256, 0, stream>>>(node_features, W, b, hbuf, out);

    // Phase 2: scatter-add edge messages into out
    edge_scatter<<<N_EDGES / 8, 256, 0, stream>>>(hbuf, edge_index, edge_type,
                                                  edge_emb, out);

    // Phase 3: ReLU
    relu_inplace<<<(N_NODES * D_OUT) / (256 * 4), 256, 0, stream>>>(out);
}